// GraphODE_14594298872164
// MI455X (gfx1250) — compile-verified
//
#include <hip/hip_runtime.h>

// GraphODE (3-layer GraphSAGE, Euler integrator) for MI455X / gfx1250.
//
//  * CSR-by-dst built once -> 30 aggregations become atomic-free coalesced
//    gathers (all working sets L2-resident on the 192MB L2).
//  * GEMMs via V_WMMA_F32_16X16X4_F32 (fp32 precision is free: ~50 GFLOP
//    total vs ~12GB of gather traffic).
//  * A-loads kept in the GLOBAL address space (no FLAT -> no DScnt/LDS-path
//    contention); B panel pre-swizzled pair-interleaved in LDS so every
//    fragment is a single conflict-free ds_load_b64.

#define N_NODES  100000
#define N_EDGES  1600000
#define DIM      64
#define T_STEPS  11
#define PSTR     160           // LDS k-pair row stride (words); 160 % 64 == 32
                               // -> the two half-waves hit disjoint bank sets

typedef __attribute__((ext_vector_type(2))) float v2f;
typedef __attribute__((ext_vector_type(8))) float v8f;

#ifndef __has_builtin
#define __has_builtin(x) 0
#endif

__device__ __forceinline__ v8f wmma_f32_4(v2f a, v2f b, v8f c) {
#if __has_builtin(__builtin_amdgcn_wmma_f32_16x16x4_f32)
  // 8 args: (neg_a, A, neg_b, B, c_mod, C, reuse_a, reuse_b)
  return __builtin_amdgcn_wmma_f32_16x16x4_f32(
      false, a, false, b, (short)0, c, false, false);
#else
  c[0] += a.x * b.x + a.y * b.y;   // keep-compiling fallback (histogram flags it)
  return c;
#endif
}

// ---------------------------------------------------------------- CSR build
__global__ void k_zero_i32(int* p, int n) {
  int i = blockIdx.x * blockDim.x + threadIdx.x;
  if (i < n) p[i] = 0;
}

__global__ void k_count(const long long* __restrict__ ei, int* __restrict__ cnt) {
  int e = blockIdx.x * blockDim.x + threadIdx.x;
  if (e < N_EDGES) {
    int dst = (int)ei[N_EDGES + e];
    atomicAdd(&cnt[dst], 1);
  }
}

// Single-block exclusive scan; also emits cursor copy and deg_inv.
__global__ void k_scan(const int* __restrict__ cnt, int* __restrict__ rowptr,
                       int* __restrict__ cursor, float* __restrict__ deg_inv,
                       int n) {
  __shared__ int sdata[256];
  __shared__ int running;
  int tid = threadIdx.x;
  if (tid == 0) running = 0;
  __syncthreads();
  for (int base = 0; base < n; base += 256) {
    int i = base + tid;
    int v = (i < n) ? cnt[i] : 0;
    sdata[tid] = v;
    __syncthreads();
    for (int off = 1; off < 256; off <<= 1) {
      int t = (tid >= off) ? sdata[tid - off] : 0;
      __syncthreads();
      sdata[tid] += t;
      __syncthreads();
    }
    int excl = sdata[tid] - v;
    if (i < n) {
      int rp = running + excl;
      rowptr[i] = rp;
      cursor[i] = rp;
      deg_inv[i] = (v > 0) ? (1.0f / (float)v) : 0.0f;
    }
    __syncthreads();
    if (tid == 0) running += sdata[255];
    __syncthreads();
  }
  if (tid == 0) rowptr[n] = running;
}

__global__ void k_fill(const long long* __restrict__ ei,
                       int* __restrict__ cursor, int* __restrict__ csr) {
  int e = blockIdx.x * blockDim.x + threadIdx.x;
  if (e < N_EDGES) {
    int src = (int)ei[e];
    int dst = (int)ei[N_EDGES + e];
    int pos = atomicAdd(&cursor[dst], 1);
    csr[pos] = src;
  }
}

__global__ void k_copy_f32(float* __restrict__ dst, const float* __restrict__ src,
                           long long n) {
  long long i = (long long)blockIdx.x * blockDim.x + threadIdx.x;
  if (i < n) dst[i] = src[i];
}

// ------------------------------------------------- mean aggregation (gather)
// One wave32 per node; lane holds dims {2*lane, 2*lane+1}. Unrolled x4 so
// four coalesced 256B row reads are in flight (covers L2 latency).
__global__ void __launch_bounds__(256)
k_aggregate(float* __restrict__ mean, const float* __restrict__ h,
            const int* __restrict__ rowptr, const int* __restrict__ csr,
            const float* __restrict__ deg_inv) {
  int node = (blockIdx.x * blockDim.x + threadIdx.x) >> 5;  // wave32
  int lane = threadIdx.x & 31;
  if (node >= N_NODES) return;
  int beg = rowptr[node], end = rowptr[node + 1];
  const int c = lane * 2;
  float ax = 0.f, ay = 0.f;
  int e = beg;
  for (; e + 4 <= end; e += 4) {
    int s0 = csr[e + 0], s1 = csr[e + 1], s2 = csr[e + 2], s3 = csr[e + 3];
    float2 v0 = *(const float2*)(h + (size_t)s0 * DIM + c);
    float2 v1 = *(const float2*)(h + (size_t)s1 * DIM + c);
    float2 v2 = *(const float2*)(h + (size_t)s2 * DIM + c);
    float2 v3 = *(const float2*)(h + (size_t)s3 * DIM + c);
    ax += v0.x + v1.x + v2.x + v3.x;
    ay += v0.y + v1.y + v2.y + v3.y;
  }
  for (; e < end; ++e) {
    int s = csr[e];
    float2 v = *(const float2*)(h + (size_t)s * DIM + c);
    ax += v.x; ay += v.y;
  }
  float di = deg_inv[node];
  float2 o = {ax * di, ay * di};
  *(float2*)(mean + (size_t)node * DIM + c) = o;
}

// --------------------------------------------------- fused SAGE layer (WMMA)
// out = epilogue( [mean | hroot](16x128) @ [Wn ; Wr](128x64) + bias )
// mode 0: ReLU ; mode 1: hbase + dt*val (layer 3 fused with Euler step).
// One wave per 16-row tile, K in steps of 4 (f32 WMMA), 4 N-tiles of 16.
__global__ void __launch_bounds__(256)
k_sage(float* __restrict__ out, const float* __restrict__ mean,
       const float* __restrict__ hroot, const float* __restrict__ Wn,
       const float* __restrict__ bias, const float* __restrict__ Wr,
       int mode, const float* __restrict__ hbase,
       const float* __restrict__ ts, int tstep) {
  // Pair-interleaved weight panel: sW[(k>>1)*PSTR + n*2 + (k&1)] = W[k][n].
  // A lane's B fragment (W[k][n], W[k+1][n]) is then ONE aligned ds_load_b64.
  __shared__ float sW[64 * PSTR];      // 64 k-pairs (K=128) x PSTR words
  __shared__ float sB[DIM];
  for (int i = threadIdx.x; i < DIM * DIM; i += blockDim.x) {
    int k = i >> 6, n = i & 63;
    int lo = (k & 1);
    sW[(k >> 1) * PSTR + n * 2 + lo]        = Wn[i];   // K rows 0..63
    sW[((k >> 1) + 32) * PSTR + n * 2 + lo] = Wr[i];   // K rows 64..127
  }
  for (int i = threadIdx.x; i < DIM; i += blockDim.x) sB[i] = bias[i];
  __syncthreads();

  const int lane = threadIdx.x & 31;
  const int tile = blockIdx.x * (blockDim.x >> 5) + (threadIdx.x >> 5);
  const int ntiles = N_NODES / 16;  // 6250, exact
  if (tile >= ntiles) return;       // wave-uniform -> EXEC all-ones for WMMA

  const int row0  = tile << 4;
  const int m     = lane & 15;      // A: M index / B,C: N index
  const int half  = lane >> 4;      // 0 or 1
  const int khalf = half << 1;      // K sub-offset {0,2} inside each 4-K step

  v8f acc[4] = {};
  // Direct kernel-arg-derived pointers -> GLOBAL (not FLAT) loads.
  const float* mrow = mean  + (size_t)(row0 + m) * DIM + khalf;
  const float* hrow = hroot + (size_t)(row0 + m) * DIM + khalf;

  // ---- segment 0: mean @ Wn  (K rows 0..63 -> pairs 0..31)
#pragma unroll
  for (int kb = 0; kb < 16; ++kb) {
    float2 av = *(const float2*)(mrow + kb * 4);
    v2f a; a.x = av.x; a.y = av.y;
    const int pbase = (kb * 2 + half) * PSTR + m * 2;
#pragma unroll
    for (int nt = 0; nt < 4; ++nt) {
      v2f b = *(const v2f*)&sW[pbase + nt * 32];
      acc[nt] = wmma_f32_4(a, b, acc[nt]);
    }
  }
  // ---- segment 1: hroot @ Wr (K rows 64..127 -> pairs 32..63)
#pragma unroll
  for (int kb = 0; kb < 16; ++kb) {
    float2 av = *(const float2*)(hrow + kb * 4);
    v2f a; a.x = av.x; a.y = av.y;
    const int pbase = (32 + kb * 2 + half) * PSTR + m * 2;
#pragma unroll
    for (int nt = 0; nt < 4; ++nt) {
      v2f b = *(const v2f*)&sW[pbase + nt * 32];
      acc[nt] = wmma_f32_4(a, b, acc[nt]);
    }
  }

  // Epilogue: C/D layout -> VGPR i, lane: row = row0 + i + 8*half, col = nt*16+m
  float dt = 1.0f;
  if (mode == 1) dt = ts[tstep] - ts[tstep - 1];
#pragma unroll
  for (int i = 0; i < 8; ++i) {
    int r = row0 + i + half * 8;
    float* orow = out + (size_t)r * DIM;
    const float* hbrow = (mode == 1) ? (hbase + (size_t)r * DIM) : nullptr;
#pragma unroll
    for (int nt = 0; nt < 4; ++nt) {
      int col = nt * 16 + m;
      float val = acc[nt][i] + sB[col];
      if (mode == 0) val = fmaxf(val, 0.0f);
      else           val = hbrow[col] + dt * val;
      orow[col] = val;
    }
  }
}

// ------------------------------------------------------- decoder projection
__global__ void k_traj(float* __restrict__ traj, const float* __restrict__ sol,
                       const float* __restrict__ Wd, const float* __restrict__ bd,
                       long long total) {
  long long i = (long long)blockIdx.x * blockDim.x + threadIdx.x;
  if (i >= total) return;
  const float* row = sol + i * DIM;
  float a0 = bd[0], a1 = bd[1];
#pragma unroll 8
  for (int k = 0; k < DIM; ++k) {
    float v = row[k];
    a0 += v * Wd[k * 2 + 0];
    a1 += v * Wd[k * 2 + 1];
  }
  traj[i * 2 + 0] = a0;
  traj[i * 2 + 1] = a1;
}

// ------------------------------------------------------------------- driver
extern "C" void kernel_launch(void* const* d_in, const int* in_sizes, int n_in,
                              void* d_out, int out_size, void* d_ws, size_t ws_size,
                              hipStream_t stream) {
  (void)in_sizes; (void)n_in; (void)out_size; (void)ws_size;
  const float*     x   = (const float*)d_in[0];
  const long long* ei  = (const long long*)d_in[1];   // int64 edge_index [2,E]
  const float*     ts  = (const float*)d_in[2];
  const float*     W1n = (const float*)d_in[3];
  const float*     b1  = (const float*)d_in[4];
  const float*     W1r = (const float*)d_in[5];
  const float*     W2n = (const float*)d_in[6];
  const float*     b2  = (const float*)d_in[7];
  const float*     W2r = (const float*)d_in[8];
  const float*     W3n = (const float*)d_in[9];
  const float*     b3  = (const float*)d_in[10];
  const float*     W3r = (const float*)d_in[11];
  const float*     Wd  = (const float*)d_in[12];
  const float*     bd  = (const float*)d_in[13];

  float* traj = (float*)d_out;
  float* sol  = traj + (size_t)T_STEPS * N_NODES * 2;   // solution slab

  char*  ws  = (char*)d_ws;
  size_t off = 0;
  auto alloc = [&](size_t bytes) -> void* {
    void* p = ws + off;
    off = (off + bytes + 255) & ~(size_t)255;
    return p;
  };
  int*   cnt     = (int*)  alloc((size_t)N_NODES * 4);
  int*   rowptr  = (int*)  alloc((size_t)(N_NODES + 1) * 4);
  int*   cursor  = (int*)  alloc((size_t)N_NODES * 4);
  float* deg_inv = (float*)alloc((size_t)N_NODES * 4);
  int*   csr     = (int*)  alloc((size_t)N_EDGES * 4);
  float* meanb   = (float*)alloc((size_t)N_NODES * DIM * 4);
  float* h1      = (float*)alloc((size_t)N_NODES * DIM * 4);
  float* h2      = (float*)alloc((size_t)N_NODES * DIM * 4);

  const int TB = 256;
  k_zero_i32<<<(N_NODES + TB - 1) / TB, TB, 0, stream>>>(cnt, N_NODES);
  k_count   <<<(N_EDGES + TB - 1) / TB, TB, 0, stream>>>(ei, cnt);
  k_scan    <<<1, TB, 0, stream>>>(cnt, rowptr, cursor, deg_inv, N_NODES);
  k_fill    <<<(N_EDGES + TB - 1) / TB, TB, 0, stream>>>(ei, cursor, csr);

  k_copy_f32<<<(int)(((size_t)N_NODES * DIM + TB - 1) / TB), TB, 0, stream>>>(
      sol, x, (long long)N_NODES * DIM);

  const int gAgg  = (N_NODES * 32 + TB - 1) / TB;          // 1 wave / node
  const int gSage = (N_NODES / 16 + 7) / 8;                // 8 tiles / block

  for (int t = 1; t < T_STEPS; ++t) {
    const float* hprev = sol + (size_t)(t - 1) * N_NODES * DIM;
    float*       hout  = sol + (size_t)t * N_NODES * DIM;

    k_aggregate<<<gAgg, TB, 0, stream>>>(meanb, hprev, rowptr, csr, deg_inv);
    k_sage<<<gSage, TB, 0, stream>>>(h1, meanb, hprev, W1n, b1, W1r,
                                     /*mode=*/0, nullptr, ts, t);
    k_aggregate<<<gAgg, TB, 0, stream>>>(meanb, h1, rowptr, csr, deg_inv);
    k_sage<<<gSage, TB, 0, stream>>>(h2, meanb, h1, W2n, b2, W2r,
                                     /*mode=*/0, nullptr, ts, t);
    k_aggregate<<<gAgg, TB, 0, stream>>>(meanb, h2, rowptr, csr, deg_inv);
    k_sage<<<gSage, TB, 0, stream>>>(hout, meanb, h2, W3n, b3, W3r,
                                     /*mode=*/1, hprev, ts, t);
  }

  long long totTN = (long long)T_STEPS * N_NODES;
  k_traj<<<(int)((totTN + TB - 1) / TB), TB, 0, stream>>>(traj, sol, Wd, bd, totTN);
}